// AttH_9036611190966
// MI455X (gfx1250) — compile-verified
//
#include <hip/hip_runtime.h>
#include <math.h>

// AttH forward for MI455X (gfx1250, wave32).
// One query per wave32; lane L owns elements [4L,4L+4) and [128+4L,128+4L+4)
// of every length-256 vector. All reductions are 5-step __shfl_xor trees.
// Persistent grid-stride waves + global_prefetch_b8 of next query's rows.
// Factor outputs (820MB write-once) use non-temporal 128-bit stores.
// The pure-copy half of rel_emb[r] goes through the CDNA5 async path:
// GLOBAL_LOAD_ASYNC_TO_LDS_B128 -> GLOBAL_STORE_ASYNC_FROM_LDS_B128,
// tracked with ASYNCcnt and overlapped with the wave's compute.

typedef float f4 __attribute__((ext_vector_type(4)));
typedef int   v4i __attribute__((ext_vector_type(4)));

#define DDIM 256
#define MIN_NORM 1e-15f
#define BALL_EPS 0.004f
#define ATT_SCALE 0.0625f   /* 1/sqrt(256) */

#if defined(__AMDGCN__) && \
    __has_builtin(__builtin_amdgcn_global_load_async_to_lds_b128) && \
    __has_builtin(__builtin_amdgcn_global_store_async_from_lds_b128)
#define HAS_ASYNC_LDS 1
#else
#define HAS_ASYNC_LDS 0
#endif

#define AS_GLOBAL __attribute__((address_space(1)))
#define AS_LDS    __attribute__((address_space(3)))

__device__ __forceinline__ void wait_asynccnt0() {
#if defined(__AMDGCN__)
#if __has_builtin(__builtin_amdgcn_s_wait_asynccnt)
    __builtin_amdgcn_s_wait_asynccnt(0);
#else
    asm volatile("s_wait_asynccnt 0" ::: "memory");
#endif
#endif
}

__device__ __forceinline__ float wave_sum32(float v) {
    v += __shfl_xor(v, 16, 32);
    v += __shfl_xor(v, 8, 32);
    v += __shfl_xor(v, 4, 32);
    v += __shfl_xor(v, 2, 32);
    v += __shfl_xor(v, 1, 32);
    return v;
}

__device__ __forceinline__ float dot4(f4 a, f4 b) {
    return a.x * b.x + a.y * b.y + a.z * b.z + a.w * b.w;
}

__device__ __forceinline__ float artanh_f(float x) {
    x = fminf(fmaxf(x, -1.0f + 1e-5f), 1.0f - 1e-5f);
    return 0.5f * (log1pf(x) - log1pf(-x));
}

// Givens rotation on two (even,odd) pairs packed in a float4.
__device__ __forceinline__ f4 givens_rot4(f4 g, f4 x) {
    float n0 = fmaxf(sqrtf(g.x * g.x + g.y * g.y), MIN_NORM);
    float n1 = fmaxf(sqrtf(g.z * g.z + g.w * g.w), MIN_NORM);
    float gx0 = g.x / n0, gy0 = g.y / n0;
    float gx1 = g.z / n1, gy1 = g.w / n1;
    f4 r;
    r.x = gx0 * x.x - gy0 * x.y;
    r.y = gx0 * x.y + gy0 * x.x;
    r.z = gx1 * x.z - gy1 * x.w;
    r.w = gx1 * x.w + gy1 * x.z;
    return r;
}

// Givens reflection on two pairs.
__device__ __forceinline__ f4 givens_ref4(f4 g, f4 x) {
    float n0 = fmaxf(sqrtf(g.x * g.x + g.y * g.y), MIN_NORM);
    float n1 = fmaxf(sqrtf(g.z * g.z + g.w * g.w), MIN_NORM);
    float gx0 = g.x / n0, gy0 = g.y / n0;
    float gx1 = g.z / n1, gy1 = g.w / n1;
    f4 r;
    r.x = gx0 * x.x + gy0 * x.y;
    r.y = gy0 * x.x - gx0 * x.y;
    r.z = gx1 * x.z + gy1 * x.w;
    r.w = gy1 * x.z - gx1 * x.w;
    return r;
}

__global__ __launch_bounds__(256) void atth_fwd_kernel(
    const int* __restrict__ queries,      // (NQ,3) int32
    const float* __restrict__ ent,        // (E,256)
    const float* __restrict__ rel_emb,    // (R,512)
    const float* __restrict__ bias_h,     // (E,1)
    const float* __restrict__ bias_t,     // (E,1)
    const float* __restrict__ c_arr,      // (R,1)
    const float* __restrict__ att_rel,    // (R,512)  [rot | ref]
    const float* __restrict__ ctx_emb,    // (R,256)
    float* __restrict__ out,              // NQ*(1+256+512+256)
    int NQ, int numWaves)
{
    const int lane = threadIdx.x & 31;
    const int wid  = threadIdx.x >> 5;   // wave within block
    const int wave = blockIdx.x * (blockDim.x >> 5) + wid;

    float* out_pred = out;
    float* out_head = out + (size_t)NQ;
    float* out_rel  = out + (size_t)NQ * 257;   // NQ + NQ*256
    float* out_tail = out + (size_t)NQ * 769;   // NQ + NQ*256 + NQ*512

#if HAS_ASYNC_LDS
    // Per-wave 1KB staging slice for the pass-through half of rel_emb[r].
    __shared__ float relbuf[8 * 256];
    AS_LDS v4i* lslice = (AS_LDS v4i*)(&relbuf[wid * 256 + lane * 4]);
#endif

    for (int q = wave; q < NQ; q += numWaves) {
        const int h = queries[3 * q + 0];
        const int r = queries[3 * q + 1];
        const int t = queries[3 * q + 2];

#if HAS_ASYNC_LDS
        // WAR: previous iteration's async store must be done with our LDS slice.
        wait_asynccnt0();
        {
            const float* rsrc = rel_emb + (size_t)r * 512 + 256 + lane * 4;
            AS_GLOBAL v4i* gsrc = (AS_GLOBAL v4i*)(const_cast<float*>(rsrc));
            // INST_OFFSET is added to both global and LDS addresses (ISA §08).
            __builtin_amdgcn_global_load_async_to_lds_b128(gsrc, lslice, 0, 0);
            __builtin_amdgcn_global_load_async_to_lds_b128(gsrc, lslice, 512, 0);
        }
#endif

        // ---- prefetch next iteration's rows (global_prefetch_b8 path) ----
        int qn = q + numWaves;
        if (qn < NQ) {
            int hn = queries[3 * qn + 0];
            int rn = queries[3 * qn + 1];
            int tn = queries[3 * qn + 2];
            const char* ph = (const char*)(ent + (size_t)hn * DDIM);
            const char* pt = (const char*)(ent + (size_t)tn * DDIM);
            const char* pr = (const char*)(rel_emb + (size_t)rn * 2 * DDIM);
            const char* pa = (const char*)(att_rel + (size_t)rn * 2 * DDIM);
            const char* pc = (const char*)(ctx_emb + (size_t)rn * DDIM);
            __builtin_prefetch(ph + lane * 32, 0, 3);  // 32 lanes x 32B = 1KB row
            __builtin_prefetch(pt + lane * 32, 0, 3);
            __builtin_prefetch(pr + lane * 64, 0, 3);  // 2KB row
            __builtin_prefetch(pa + lane * 64, 0, 3);
            __builtin_prefetch(pc + lane * 32, 0, 3);
        }

        // ---- gathers: 128-bit vector loads, lane L -> elems 4L and 128+4L ----
        const f4* Eh = (const f4*)(ent + (size_t)h * DDIM);
        const f4* Et = (const f4*)(ent + (size_t)t * DDIM);
        const f4* Ar = (const f4*)(att_rel + (size_t)r * 2 * DDIM);
        const f4* Cx = (const f4*)(ctx_emb + (size_t)r * DDIM);
        const f4* Rl = (const f4*)(rel_emb + (size_t)r * 2 * DDIM);

        f4 h0 = Eh[lane],       h1 = Eh[32 + lane];
        f4 ga0 = Ar[lane],      ga1 = Ar[32 + lane];     // rot_mat
        f4 gb0 = Ar[64 + lane], gb1 = Ar[96 + lane];     // ref_mat
        f4 cx0 = Cx[lane],      cx1 = Cx[32 + lane];
        f4 rl0 = Rl[lane],      rl1 = Rl[32 + lane];     // rel (first half)
#if !HAS_ASYNC_LDS
        f4 rl2 = Rl[64 + lane], rl3 = Rl[96 + lane];     // rel (copy-only half)
#endif
        f4 t0 = Et[lane],       t1 = Et[32 + lane];

        // ---- c_q = softplus(c[r]) ----
        float craw = c_arr[r];
        float cq = (craw > 20.0f) ? craw : log1pf(expf(craw));
        float sc = sqrtf(cq);
        float maxnorm = (1.0f - BALL_EPS) / sc;

        // ---- Givens rotation / reflection ----
        f4 rot0 = givens_rot4(ga0, h0), rot1 = givens_rot4(ga1, h1);
        f4 ref0 = givens_ref4(gb0, h0), ref1 = givens_ref4(gb1, h1);

        // ---- attention: softmax over {ref, rot} ----
        float s_ref = wave_sum32(dot4(cx0, ref0) + dot4(cx1, ref1)) * ATT_SCALE;
        float s_rot = wave_sum32(dot4(cx0, rot0) + dot4(cx1, rot1)) * ATT_SCALE;
        float m = fmaxf(s_ref, s_rot);
        float w_ref = expf(s_ref - m);
        float w_rot = expf(s_rot - m);
        float winv = 1.0f / (w_ref + w_rot);
        w_ref *= winv; w_rot *= winv;
        f4 att0 = w_ref * ref0 + w_rot * rot0;
        f4 att1 = w_ref * ref1 + w_rot * rot1;

        // ---- lhs = project(expmap0(att, c)) ----
        float un2 = wave_sum32(dot4(att0, att0) + dot4(att1, att1));
        float un = fmaxf(sqrtf(un2), MIN_NORM);
        float kx = tanhf(sc * un) / (sc * un);
        float gn = fmaxf(un * kx, MIN_NORM);           // = |expmap0(att)|
        if (gn > maxnorm) kx *= maxnorm / gn;
        f4 x0 = kx * att0, x1 = kx * att1;

        // ---- rel' = project(expmap0(rel, c)) ----
        float rn2r = wave_sum32(dot4(rl0, rl0) + dot4(rl1, rl1));
        float rnr = fmaxf(sqrtf(rn2r), MIN_NORM);
        float kr = tanhf(sc * rnr) / (sc * rnr);
        float gr = fmaxf(rnr * kr, MIN_NORM);
        if (gr > maxnorm) kr *= maxnorm / gr;
        f4 y0 = kr * rl0, y1 = kr * rl1;

        // ---- res = project(mobius_add(lhs, rel', c)) ----
        float x2 = wave_sum32(dot4(x0, x0) + dot4(x1, x1));
        float y2 = wave_sum32(dot4(y0, y0) + dot4(y1, y1));
        float xy = wave_sum32(dot4(x0, y0) + dot4(x1, y1));
        float A = 1.0f + 2.0f * cq * xy + cq * y2;
        float B = 1.0f - cq * x2;
        float den = 1.0f + 2.0f * cq * xy + cq * cq * x2 * y2;
        float dinv = 1.0f / fmaxf(den, MIN_NORM);
        f4 e0 = (A * x0 + B * y0) * dinv;
        f4 e1 = (A * x1 + B * y1) * dinv;
        float pn2 = wave_sum32(dot4(e0, e0) + dot4(e1, e1));
        float pn = fmaxf(sqrtf(pn2), MIN_NORM);
        float s = (pn > maxnorm) ? (maxnorm / pn) : 1.0f;
        e0 *= s; e1 *= s;
        float x2h = pn2 * s * s;                       // |res|^2

        // ---- dist^2 = hyp_distance(res, tail, c)^2 ----
        float tn2 = wave_sum32(dot4(t0, t0) + dot4(t1, t1));
        float vnorm = fmaxf(sqrtf(tn2), MIN_NORM);
        float xv = wave_sum32(dot4(e0, t0) + dot4(e1, t1)) / vnorm;
        float gamma = tanhf(sc * vnorm) / vnorm;
        float c1 = 1.0f - 2.0f * cq * gamma * xv + cq * gamma * gamma;
        float c2 = 1.0f - cq * x2h;
        float numd = sqrtf(fmaxf(c1 * c1 * x2h + c2 * c2 * gamma * gamma
                                 - 2.0f * c1 * c2 * gamma * xv, 0.0f));
        float dend = fmaxf(1.0f - 2.0f * cq * gamma * xv
                           + cq * cq * gamma * gamma * x2h, MIN_NORM);
        float dist = 2.0f * artanh_f(sc * numd / dend) / sc;

        if (lane == 0)
            out_pred[q] = bias_h[h] + bias_t[t] + dist * dist;

        // ---- factor outputs: non-temporal 128-bit streaming stores ----
        f4* OH = (f4*)(out_head + (size_t)q * 256);
        __builtin_nontemporal_store(h0, OH + lane);
        __builtin_nontemporal_store(h1, OH + 32 + lane);
        f4* OR = (f4*)(out_rel + (size_t)q * 512);
        __builtin_nontemporal_store(rl0, OR + lane);
        __builtin_nontemporal_store(rl1, OR + 32 + lane);
#if HAS_ASYNC_LDS
        {
            // RAW: async loads must have landed in LDS before the async store reads it.
            wait_asynccnt0();
            float* rdst = out_rel + (size_t)q * 512 + 256 + lane * 4;
            AS_GLOBAL v4i* gdst = (AS_GLOBAL v4i*)(rdst);
            __builtin_amdgcn_global_store_async_from_lds_b128(gdst, lslice, 0, 0);
            __builtin_amdgcn_global_store_async_from_lds_b128(gdst, lslice, 512, 0);
        }
#else
        __builtin_nontemporal_store(rl2, OR + 64 + lane);
        __builtin_nontemporal_store(rl3, OR + 96 + lane);
#endif
        f4* OT = (f4*)(out_tail + (size_t)q * 256);
        __builtin_nontemporal_store(t0, OT + lane);
        __builtin_nontemporal_store(t1, OT + 32 + lane);
    }
    // S_ENDPGM performs an implicit wait-idle, covering outstanding async stores.
}

extern "C" void kernel_launch(void* const* d_in, const int* in_sizes, int n_in,
                              void* d_out, int out_size, void* d_ws, size_t ws_size,
                              hipStream_t stream) {
    const int*   queries = (const int*)d_in[0];
    const float* ent     = (const float*)d_in[1];
    const float* rel     = (const float*)d_in[2];
    const float* bh      = (const float*)d_in[3];
    const float* bt      = (const float*)d_in[4];
    const float* c       = (const float*)d_in[5];
    const float* ar      = (const float*)d_in[6];
    const float* cx      = (const float*)d_in[7];

    const int NQ = in_sizes[0] / 3;
    const int threads = 256;                 // 8 wave32s per block
    const int blocks = 1024;                 // persistent grid-stride
    const int numWaves = blocks * (threads / 32);

    atth_fwd_kernel<<<blocks, threads, 0, stream>>>(
        queries, ent, rel, bh, bt, c, ar, cx, (float*)d_out, NQ, numWaves);
}